// LaStGaitAdapter_51299089383909
// MI455X (gfx1250) — compile-verified
//
#include <hip/hip_runtime.h>
#include <hip/hip_bf16.h>
#include <math.h>
#include <stdint.h>

typedef float v2f __attribute__((ext_vector_type(2)));
typedef float v4f __attribute__((ext_vector_type(4)));
typedef float v8f __attribute__((ext_vector_type(8)));
typedef unsigned int u32x4 __attribute__((ext_vector_type(4)));
typedef int i32x8 __attribute__((ext_vector_type(8)));
typedef int i32x4 __attribute__((ext_vector_type(4)));

#define N_    8
#define C_    256
#define S_    64
#define H_    32
#define Wd_   16
#define BINS_ 4
#define T_    (S_*BINS_)      /* 256 tokens per batch */
#define M_    (N_*T_)         /* 2048 GEMM rows       */
#define KSEL_ 90              /* round(256*0.35)      */
#define EPS_  1e-6f

// ---------------------------------------------------------------------------
// Kernel 1: x[n,c,s,h,w] -> tokens[(n*T + s*4+bin), c]  (mean over 8h x 16w)
// block = 128 threads = 4 waves, one block per (n,c,s); each wave owns 1 bin.
// ---------------------------------------------------------------------------
__global__ void tok_kernel(const float* __restrict__ x, float* __restrict__ tok) {
    const int b   = blockIdx.x;               // n*C_*S_ blocks
    const int s   = b % S_;
    const int c   = (b / S_) % C_;
    const int n   = b / (S_ * C_);
    const int tid = threadIdx.x;              // 0..127
    const size_t base = ((size_t)(n*C_ + c)*S_ + s) * (H_*Wd_);   // 512 floats
    const v4f v = *reinterpret_cast<const v4f*>(x + base + tid*4);
    float sum = v.x + v.y + v.z + v.w;
    #pragma unroll
    for (int off = 16; off > 0; off >>= 1)
        sum += __shfl_down(sum, off, 32);
    if ((tid & 31) == 0) {
        const int bin = tid >> 5;             // wave id == bin (128 floats/bin)
        tok[(size_t)(n*T_ + s*BINS_ + bin)*C_ + c] = sum * (1.0f/128.0f);
    }
}

// ---------------------------------------------------------------------------
// Kernel 2: kk[d] = irfft(gaussian mask)  (ortho norms cancel -> plain conv)
// ---------------------------------------------------------------------------
__global__ void kk_kernel(float* __restrict__ kk) {
    const int d = threadIdx.x;                // 0..255
    float acc = 1.0f;                         // mask[0] = 1
    for (int f = 1; f < 128; ++f) {
        const float ff = (float)f * (1.0f/32.0f);       // (f/128)/0.25
        const float m  = __expf(-0.5f * ff * ff);
        acc += 2.0f * m * __cosf(6.28318530717958647692f * (float)(f*d) * (1.0f/256.0f));
    }
    acc += __expf(-8.0f) * __cosf(3.14159265358979323846f * (float)d);   // f=128 term
    kk[d] = acc * (1.0f/256.0f);
}

__global__ void wmat_kernel(const float* __restrict__ kk, float* __restrict__ W) {
    const int i = blockIdx.x, j = threadIdx.x;
    W[i*C_ + j] = kk[(j - i) & (C_-1)];
}

__global__ void zero_kernel(float* __restrict__ p, int n) {
    const int i = blockIdx.x*blockDim.x + threadIdx.x;
    if (i < n) p[i] = 0.0f;
}

// ---------------------------------------------------------------------------
// Kernel 3: low[m][j] = sum_i tok[m][i] * W[i][j]   via v_wmma_f32_16x16x4_f32
// grid (16 mblocks, 16 ntiles), 256 thr = 8 waves; wave -> one 16x16 D tile.
// B tile (W[:, n0:n0+16], 256x16 f32) staged into LDS by the Tensor Data
// Mover: 2D D# with tile_dim0=16 (contiguous), tile_dim1=256, stride=256.
// ---------------------------------------------------------------------------
__global__ void gemm_kernel(const float* __restrict__ tok,
                            const float* __restrict__ W,
                            float* __restrict__ low) {
    __shared__ float bs[256*16];              // row-major [k][nn], TDM layout
    const int tid  = threadIdx.x;
    const int n0   = blockIdx.y * 16;

#if __has_builtin(__builtin_amdgcn_tensor_load_to_lds) && __has_builtin(__builtin_amdgcn_s_wait_tensorcnt)
    if (tid < 32) {                           // wave 0 issues one TDM op
        const unsigned lds_off = (unsigned)(uintptr_t)(void*)bs;   // LDS byte offset
        const unsigned long long ga = (unsigned long long)(uintptr_t)(W + n0);
        u32x4 g0;
        g0.x = 1u;                                            // count=1 (valid), user mode
        g0.y = lds_off;                                       // D#.lds_addr
        g0.z = (unsigned)(ga & 0xFFFFFFFFull);                // global_addr[31:0]
        g0.w = ((unsigned)(ga >> 32) & 0x01FFFFFFu)           // global_addr[56:32]
               | (2u << 30);                                  // type=2 ("image")
        i32x8 g1;
        g1[0] = 0x00020000;            // wg_mask=0, data_size=2 (4B), no pad/iter
        g1[1] = 0;                     // atomic_barrier=0, tensor_dim0[15:0]=0
        g1[2] = 0x0010;                // tensor_dim0 = 1M (no OOB clip)
        g1[3] = 0x0010 | (16 << 16);   // tensor_dim1 = 1M ; tile_dim0 = 16
        g1[4] = 256;                   // tile_dim1 = 256 rows ; tile_dim2 = 0
        g1[5] = 256;                   // tensor_dim0_stride = 256 elements
        g1[6] = 0;                     // stride hi / tensor_dim1_stride lo
        g1[7] = 0;                     // tensor_dim1_stride hi
        i32x4 gz4 = {};                // groups 2/3 unused (2D tile)
        i32x8 gz8 = {};                // extended group (unused)
        __builtin_amdgcn_tensor_load_to_lds(g0, g1, gz4, gz4, gz8, 0);
        __builtin_amdgcn_s_wait_tensorcnt(0);
    }
#else
    {
        const int nn = tid & 15, kb = (tid >> 4) * 16;
        #pragma unroll
        for (int q = 0; q < 16; ++q)
            bs[(kb + q)*16 + nn] = W[(kb + q)*C_ + n0 + nn];
    }
#endif
    __syncthreads();

    const int wid  = tid >> 5;
    const int lane = tid & 31;
    const int hh   = lane >> 4;               // lane half
    const int mm   = lane & 15;
    const int m0   = blockIdx.x * 128 + wid * 16;

    v8f acc = {};
    const float* arow = tok + (size_t)(m0 + mm) * C_;
    #pragma unroll 4
    for (int k0 = 0; k0 < C_; k0 += 4) {
        const v2f a = *reinterpret_cast<const v2f*>(arow + k0 + 2*hh);
        v2f bv;
        bv.x = bs[(k0 + 2*hh    )*16 + mm];
        bv.y = bs[(k0 + 2*hh + 1)*16 + mm];
        acc = __builtin_amdgcn_wmma_f32_16x16x4_f32(
                  false, a, false, bv, (short)0, acc, false, false);
    }
    #pragma unroll
    for (int r = 0; r < 8; ++r)
        low[(size_t)(m0 + r + 8*hh)*C_ + n0 + mm] = acc[r];
}

// ---------------------------------------------------------------------------
// Kernel 4: stability + exact top-k membership vote.
// block per (n,c); thread t: rank of st[t] in row; selected iff rank < KSEL.
// ---------------------------------------------------------------------------
__global__ void vote_kernel(const float* __restrict__ tok,
                            const float* __restrict__ low,
                            float* __restrict__ vote) {
    __shared__ float st[T_];
    const int b = blockIdx.x;                 // n*C_ blocks
    const int c = b & (C_-1);
    const int n = b >> 8;
    const int t = threadIdx.x;
    const size_t idx = (size_t)(n*T_ + t)*C_ + c;
    const float lo = low[idx];
    const float my = fabsf(lo) / (fabsf(lo - tok[idx]) + EPS_);
    st[t] = my;
    __syncthreads();
    int rank = 0;
    #pragma unroll 8
    for (int t2 = 0; t2 < T_; ++t2) {
        const float v = st[t2];
        rank += (v > my) || (v == my && t2 < t);
    }
    if (rank < KSEL_) atomicAdd(&vote[n*T_ + t], 1.0f);
}

// ---------------------------------------------------------------------------
// Kernel 5: normalize vote, apply tanh gate strength, clamp -> gate_tok[n][t]
// ---------------------------------------------------------------------------
__global__ void gate_kernel(const float* __restrict__ vote,
                            const float* __restrict__ gstr,
                            float* __restrict__ gate_tok) {
    __shared__ float red[T_];
    const int n = blockIdx.x, t = threadIdx.x;
    const float v = vote[n*T_ + t] * (1.0f/(float)C_);
    red[t] = v;
    __syncthreads();
    #pragma unroll
    for (int off = 128; off > 0; off >>= 1) {
        if (t < off) red[t] += red[t + off];
        __syncthreads();
    }
    const float mean  = red[0] * (1.0f/(float)T_);
    const float denom = fmaxf(mean, EPS_);
    const float gs    = tanhf(gstr[0]);
    float g = 1.0f + gs * (v/denom - 1.0f);
    g = fminf(fmaxf(g, 0.75f), 1.25f);
    gate_tok[n*T_ + t] = g;
}

// ---------------------------------------------------------------------------
// Kernel 6: fused gate-interp multiply + delta-residual interp add.
// thread per (n,c,s,h) row of 16 w-elements; wave reads 2KB contiguous.
// ---------------------------------------------------------------------------
__global__ void out_kernel(const float* __restrict__ x,
                           const float* __restrict__ tok,
                           const float* __restrict__ low,
                           const float* __restrict__ gate_tok,
                           const float* __restrict__ res_scale,
                           float* __restrict__ out) {
    const int ridx = blockIdx.x*blockDim.x + threadIdx.x;   // n*c*s*h rows
    const int h = ridx & (H_-1);
    const int s = (ridx >> 5) & (S_-1);
    const int c = (ridx >> 11) & (C_-1);
    const int n = ridx >> 19;

    // bilinear (align_corners=False) bins(4) -> h(32): src = (h+.5)/8 - .5
    float src = ((float)h + 0.5f) * 0.125f - 0.5f;
    src = fminf(fmaxf(src, 0.0f), 3.0f);
    const int   i0 = (int)src;
    const int   i1 = min(i0 + 1, BINS_-1);
    const float wl = src - (float)i0;

    const int t0 = s*BINS_ + i0, t1 = s*BINS_ + i1;
    const float g = gate_tok[n*T_ + t0]*(1.0f-wl) + gate_tok[n*T_ + t1]*wl;

    const size_t b0 = (size_t)(n*T_ + t0)*C_ + c;
    const size_t b1 = (size_t)(n*T_ + t1)*C_ + c;
    const float d0 = low[b0] - tok[b0];
    const float d1 = low[b1] - tok[b1];
    const float add = res_scale[c] * (d0*(1.0f-wl) + d1*wl);

    const size_t base = (size_t)ridx * Wd_;
    __builtin_prefetch(x + base + 2048, 0, 1);              // global_prefetch_b8
    #pragma unroll
    for (int q = 0; q < 4; ++q) {
        const v4f xv = *reinterpret_cast<const v4f*>(x + base + q*4);
        const v4f ov = xv * g + add;
        *reinterpret_cast<v4f*>(out + base + q*4) = ov;
    }
}

// ---------------------------------------------------------------------------
extern "C" void kernel_launch(void* const* d_in, const int* in_sizes, int n_in,
                              void* d_out, int out_size, void* d_ws, size_t ws_size,
                              hipStream_t stream) {
    const float* x    = (const float*)d_in[0];
    const float* gstr = (const float*)d_in[1];
    const float* rsc  = (const float*)d_in[2];
    float* out = (float*)d_out;

    char* ws = (char*)d_ws;
    float* tok   = (float*)(ws);                       // 2 MB
    float* low   = (float*)(ws + 2097152);             // 2 MB
    float* Wm    = (float*)(ws + 4194304);             // 256 KB
    float* kk    = (float*)(ws + 4456448);             // 1 KB
    float* vote  = (float*)(ws + 4457472);             // 8 KB
    float* gate  = (float*)(ws + 4465664);             // 8 KB

    tok_kernel <<<N_*C_*S_, 128, 0, stream>>>(x, tok);
    kk_kernel  <<<1,        256, 0, stream>>>(kk);
    wmat_kernel<<<C_,       256, 0, stream>>>(kk, Wm);
    zero_kernel<<<(N_*T_+255)/256, 256, 0, stream>>>(vote, N_*T_);
    gemm_kernel<<<dim3(M_/128, C_/16), 256, 0, stream>>>(tok, Wm, low);
    vote_kernel<<<N_*C_, T_, 0, stream>>>(tok, low, vote);
    gate_kernel<<<N_,    T_, 0, stream>>>(vote, gstr, gate);
    out_kernel <<<(N_*C_*S_*H_)/256, 256, 0, stream>>>(x, tok, low, gate, rsc, out);
}